// ScaledDotProductAttention_1580547971766
// MI455X (gfx1250) — compile-verified
//
#include <hip/hip_runtime.h>
#include <math.h>

#define NB 4
#define NH 16
#define NS 2048
#define ND 64
#define NEGV -1000.0f
#define SCALE 0.125f   // 1/sqrt(64)

typedef __attribute__((ext_vector_type(2))) float v2f;
typedef __attribute__((ext_vector_type(4))) float v4f;
typedef __attribute__((ext_vector_type(8))) float v8f;

// LDS padded strides (floats) chosen for conflict-free fragment reads on 64 banks
#define KSTR 68   // K tile: 16 x 64, v2f reads at n*68 + 4k + 2h -> banks 4n(+2h) distinct
#define VSTR 72   // V tile: 16 x 64, scalar reads row*72 + col   -> halves 16 banks apart
#define PSTR 20   // P tile: 16 x 16, write C-layout / read A-layout both conflict-free

#define NT (NS / 16)   // 128 key tiles

__launch_bounds__(256)
__global__ void sdpa_f32_wmma_kernel(const float* __restrict__ Qg,
                                     const float* __restrict__ Kg,
                                     const float* __restrict__ Vg,
                                     const int*   __restrict__ Mg,
                                     float* __restrict__ Og,
                                     float* __restrict__ Ag)
{
    __shared__ float sK[2][16 * KSTR];   // double-buffered K tile
    __shared__ float sV[2][16 * VSTR];   // double-buffered V tile
    __shared__ float sP[8][16 * PSTR];   // per-wave P transpose staging

    const int tid  = threadIdx.x;
    const int wave = tid >> 5;
    const int lane = tid & 31;
    const int n    = lane & 15;   // B/C/D column index; A-matrix row index
    const int h    = lane >> 4;   // lane half selects K-pair for A/B fragments

    const int bh = blockIdx.y;                       // fused (b,h): 0..63
    const int q0 = (blockIdx.x * 8 + wave) * 16;     // this wave's q-row base

    const float* Q = Qg + (size_t)bh * NS * ND + (size_t)q0 * ND;
    const float* K = Kg + (size_t)bh * NS * ND;
    const float* V = Vg + (size_t)bh * NS * ND;
    const int*   M = Mg + (size_t)bh * NS;
    float* O = Og + (size_t)bh * NS * ND + (size_t)q0 * ND;
    float* A = Ag + (size_t)bh * NS * NS + (size_t)q0 * NS;

    // cooperative-load geometry: 256 threads x float4 covers one 16x64 tile
    const int lrow = tid >> 4;
    const int lc4  = (tid & 15) * 4;
    const float* Kld = K + (size_t)lrow * ND + lc4;
    const float* Vld = V + (size_t)lrow * ND + lc4;
    const int    ksl = lrow * KSTR + lc4;   // LDS slot (K)
    const int    vsl = lrow * VSTR + lc4;   // LDS slot (V)

    // ---- Q as A-operand fragments (16 chunks of K=4), scale folded in ----
    // A layout (ISA 16x4 f32): lane = M (0..15) in both halves; k = 2*half + vgpr
    v2f qa[16];
#pragma unroll
    for (int kk = 0; kk < 16; ++kk) {
        v2f t = *(const v2f*)(Q + n * ND + kk * 4 + 2 * h);
        v2f s; s.x = t.x * SCALE; s.y = t.y * SCALE;
        qa[kk] = s;
    }

    // =================== Pass A: online softmax stats ===================
    float rm[8], rs[8];
#pragma unroll
    for (int r = 0; r < 8; ++r) { rm[r] = -INFINITY; rs[r] = 0.0f; }

    // pipeline prologue: tile 0 into buffer 0
    {
        v4f kreg = *(const v4f*)(Kld);
        *(v4f*)(&sK[0][ksl]) = kreg;
    }
    __syncthreads();

    for (int kt = 0; kt < NT; ++kt) {
        const int key0 = kt * 16;
        const int nxt  = (kt + 1) & (NT - 1);        // wrap to avoid OOB on last iter
        // (1) issue next tile's global load early (latency overlaps compute)
        v4f kreg = *(const v4f*)(Kld + (size_t)(nxt * 16) * ND);

        // (2) compute scores from current buffer
        const float* sKc = sK[kt & 1];
        v8f acc = {};
#pragma unroll
        for (int kk = 0; kk < 16; ++kk) {
            // B layout: lane = N (key row), k = 2*half + vgpr -> K[n][4kk+2h+v]
            v2f b = *(const v2f*)(sKc + n * KSTR + kk * 4 + 2 * h);
            acc = __builtin_amdgcn_wmma_f32_16x16x4_f32(
                      false, qa[kk], false, b, (short)0, acc, false, false);
        }
        const int mkv = M[key0 + n];
#pragma unroll
        for (int r = 0; r < 8; ++r) {
            float x  = (mkv == 0) ? NEGV : acc[r];
            float nm = fmaxf(rm[r], x);
            rs[r] = rs[r] * __expf(rm[r] - nm) + __expf(x - nm);
            rm[r] = nm;
        }

        // (3) stage next tile into the other buffer; (4) one barrier/iteration
        *(v4f*)(&sK[(kt + 1) & 1][ksl]) = kreg;
        __syncthreads();
    }

    // combine stats across the 16 lanes that share the same rows (within half)
#pragma unroll
    for (int off = 1; off < 16; off <<= 1) {
#pragma unroll
        for (int r = 0; r < 8; ++r) {
            float om = __shfl_xor(rm[r], off, 32);
            float os = __shfl_xor(rs[r], off, 32);
            float nm = fmaxf(rm[r], om);
            rs[r] = rs[r] * __expf(rm[r] - nm) + os * __expf(om - nm);
            rm[r] = nm;
        }
    }
    float rinv[8];
#pragma unroll
    for (int r = 0; r < 8; ++r) rinv[r] = 1.0f / rs[r];

    // ============ Pass B: probabilities -> attn store + PV ==============
    v8f o0 = {}, o1 = {}, o2 = {}, o3 = {};
    float* sPw = sP[wave];

    // pipeline prologue: tile 0 (K and V) into buffer 0
    {
        v4f kreg = *(const v4f*)(Kld);
        v4f vreg = *(const v4f*)(Vld);
        *(v4f*)(&sK[0][ksl]) = kreg;
        *(v4f*)(&sV[0][vsl]) = vreg;
    }
    __syncthreads();

    for (int kt = 0; kt < NT; ++kt) {
        const int key0 = kt * 16;
        const int nxt  = (kt + 1) & (NT - 1);
        // (1) issue next tile's global loads early
        v4f kreg = *(const v4f*)(Kld + (size_t)(nxt * 16) * ND);
        v4f vreg = *(const v4f*)(Vld + (size_t)(nxt * 16) * ND);

        // (2) compute from current buffers
        const float* sKc = sK[kt & 1];
        const float* sVc = sV[kt & 1];

        v8f acc = {};
#pragma unroll
        for (int kk = 0; kk < 16; ++kk) {
            v2f b = *(const v2f*)(sKc + n * KSTR + kk * 4 + 2 * h);
            acc = __builtin_amdgcn_wmma_f32_16x16x4_f32(
                      false, qa[kk], false, b, (short)0, acc, false, false);
        }
        const int mkv = M[key0 + n];
#pragma unroll
        for (int r = 0; r < 8; ++r) {
            float x = (mkv == 0) ? NEGV : acc[r];
            float p = __expf(x - rm[r]) * rinv[r];
            // C-layout element: row = r + 8*half, col = n
            A[(size_t)(r + 8 * h) * NS + key0 + n] = p;      // attn output
            sPw[(r + 8 * h) * PSTR + n] = p;                 // stage for A-frags
        }
        // same-wave LDS RAW: DS ops in-order per wave; compiler orders
        // may-aliasing accesses and inserts s_wait_dscnt before the loads.

        // PV: O(16x64) += P(16x16) * V(16x64); K-dim split into 4 chunks of 4
#pragma unroll
        for (int c = 0; c < 4; ++c) {
            // A-fragment of P: row m = n, k = 4c + 2h + v (8B aligned)
            v2f a = *(const v2f*)(sPw + n * PSTR + c * 4 + 2 * h);
            const int r0 = (c * 4 + 2 * h) * VSTR;
            v2f b;
            b.x = sVc[r0 +        0 * 16 + n];
            b.y = sVc[r0 + VSTR + 0 * 16 + n];
            o0 = __builtin_amdgcn_wmma_f32_16x16x4_f32(false, a, false, b, (short)0, o0, false, false);
            b.x = sVc[r0 +        1 * 16 + n];
            b.y = sVc[r0 + VSTR + 1 * 16 + n];
            o1 = __builtin_amdgcn_wmma_f32_16x16x4_f32(false, a, false, b, (short)0, o1, false, false);
            b.x = sVc[r0 +        2 * 16 + n];
            b.y = sVc[r0 + VSTR + 2 * 16 + n];
            o2 = __builtin_amdgcn_wmma_f32_16x16x4_f32(false, a, false, b, (short)0, o2, false, false);
            b.x = sVc[r0 +        3 * 16 + n];
            b.y = sVc[r0 + VSTR + 3 * 16 + n];
            o3 = __builtin_amdgcn_wmma_f32_16x16x4_f32(false, a, false, b, (short)0, o3, false, false);
        }

        // (3) stage next tile into the other buffers; (4) one barrier/iteration
        *(v4f*)(&sK[(kt + 1) & 1][ksl]) = kreg;
        *(v4f*)(&sV[(kt + 1) & 1][vsl]) = vreg;
        __syncthreads();
    }

    // ---- write output tile (C-layout: row = r + 8h, col = 16*nb + n) ----
#pragma unroll
    for (int r = 0; r < 8; ++r) {
        const size_t m = (size_t)(r + 8 * h);
        O[m * ND +  0 + n] = o0[r];
        O[m * ND + 16 + n] = o1[r];
        O[m * ND + 32 + n] = o2[r];
        O[m * ND + 48 + n] = o3[r];
    }
}

extern "C" void kernel_launch(void* const* d_in, const int* in_sizes, int n_in,
                              void* d_out, int out_size, void* d_ws, size_t ws_size,
                              hipStream_t stream) {
    const float* q = (const float*)d_in[0];
    const float* k = (const float*)d_in[1];
    const float* v = (const float*)d_in[2];
    const int*   m = (const int*)d_in[3];
    float* out  = (float*)d_out;                                // (B,H,S,D)
    float* attn = out + (size_t)NB * NH * NS * ND;              // (B,H,S,S)

    dim3 grid(NS / 128, NB * NH);   // 8 waves/block x 16 q-rows/wave = 128 q-rows/block
    sdpa_f32_wmma_kernel<<<grid, dim3(256), 0, stream>>>(q, k, v, m, out, attn);
}